// EncoderSpin_13211319403150
// MI455X (gfx1250) — compile-verified
//
#include <hip/hip_runtime.h>

// EncoderSpin: 3x GraphConv(+ReLU) -> GraphNorm -> GraphConv mu / GraphConv logvar
// N=500000 nodes, E=2000000 edges, G=256 graphs, fp32 throughout.
// Dense matmuls use V_WMMA_F32_16X16X4_F32 (fp32-precise CDNA5 matrix path).

#define NNODES 500000
#define NEDGES 2000000
#define NGRAPH 256
#define GN_EPS 1e-5f

typedef __attribute__((ext_vector_type(2))) float v2f;
typedef __attribute__((ext_vector_type(8))) float v8f;

// ---------------------------------------------------------------- fill zeros
__global__ void fill0_kernel(float* __restrict__ p, int n) {
    int i = blockIdx.x * blockDim.x + threadIdx.x;
    if (i < n) p[i] = 0.0f;
}

// -------------------------------------------- edge gather * ew, scatter-add
// one thread per (edge, channel); consecutive threads cover consecutive
// channels of one edge -> coalesced gather of xin[src*C ..] and coalesced
// atomic adds into agg[dst*C ..] (global_atomic_add_f32, no return).
template <int C>
__global__ void edge_scatter_kernel(const float* __restrict__ xin,
                                    const int* __restrict__ src,
                                    const int* __restrict__ dst,
                                    const float* __restrict__ ew,
                                    float* __restrict__ agg, int total) {
    int i = blockIdx.x * blockDim.x + threadIdx.x;
    if (i >= total) return;
    int e = i / C;
    int c = i % C;
    float m = xin[src[e] * C + c] * ew[e];
    atomicAdd(&agg[dst[e] * C + c], m);
}

// ------------------------------------------------- layer 1 (C_in = 1, VALU)
__global__ void conv1_node_kernel(const float* __restrict__ agg1,
                                  const float* __restrict__ x,
                                  const float* __restrict__ w_rel,   // [1,16]
                                  const float* __restrict__ b_rel,   // [16]
                                  const float* __restrict__ w_root,  // [1,16]
                                  float* __restrict__ h1) {
    int i = blockIdx.x * blockDim.x + threadIdx.x; // over N*16
    if (i >= NNODES * 16) return;
    int n = i >> 4, c = i & 15;
    float v = agg1[n] * w_rel[c] + b_rel[c] + x[n] * w_root[c];
    h1[i] = fmaxf(v, 0.0f);
}

// ---------------------------- node update via WMMA: agg@Wrel + x@Wroot + b
// One wave per 16-row tile (N = 500000 = 31250 * 16, no tail).
// f32 WMMA fragment layouts (ISA 7.12.2):
//   A 16x4 : lane(l): M = l%16,  K = k0 + 2*(l>=16) + {vgpr 0,1}
//   B 4x16 : lane(l): N = l%16,  K = k0 + 2*(l>=16) + {vgpr 0,1}
//   C/D    : lane(l): N = l%16,  M = vgpr + 8*(l>=16)
template <int CIN, int COUT, bool RELU>
__global__ void conv_node_wmma_kernel(const float* __restrict__ agg,
                                      const float* __restrict__ xin,
                                      const float* __restrict__ w_rel,  // [CIN,COUT]
                                      const float* __restrict__ w_root, // [CIN,COUT]
                                      const float* __restrict__ b_rel,  // [COUT]
                                      float* __restrict__ out, int ntiles) {
    int wave = (blockIdx.x * blockDim.x + threadIdx.x) >> 5;
    if (wave >= ntiles) return;          // wave-uniform exit: EXEC all-1s below
    int lane = threadIdx.x & 31;
    int m  = lane & 15;                  // A row / B,C column
    int hi = lane >> 4;                  // lane-half selector
    int row0 = wave * 16;

#pragma unroll
    for (int nt = 0; nt < COUT / 16; ++nt) {
        v8f acc = {};
#pragma unroll
        for (int k0 = 0; k0 < CIN; k0 += 4) {       // agg @ w_rel
            int ka = k0 + 2 * hi;
            v2f a, b;
            a.x = agg[(row0 + m) * CIN + ka];
            a.y = agg[(row0 + m) * CIN + ka + 1];
            b.x = w_rel[(ka) * COUT + nt * 16 + m];
            b.y = w_rel[(ka + 1) * COUT + nt * 16 + m];
            acc = __builtin_amdgcn_wmma_f32_16x16x4_f32(
                false, a, false, b, (short)0, acc, false, false);
        }
#pragma unroll
        for (int k0 = 0; k0 < CIN; k0 += 4) {       // xin @ w_root
            int ka = k0 + 2 * hi;
            v2f a, b;
            a.x = xin[(row0 + m) * CIN + ka];
            a.y = xin[(row0 + m) * CIN + ka + 1];
            b.x = w_root[(ka) * COUT + nt * 16 + m];
            b.y = w_root[(ka + 1) * COUT + nt * 16 + m];
            acc = __builtin_amdgcn_wmma_f32_16x16x4_f32(
                false, a, false, b, (short)0, acc, false, false);
        }
#pragma unroll
        for (int v = 0; v < 8; ++v) {
            int row = row0 + v + 8 * hi;
            int col = nt * 16 + m;
            float val = acc[v] + b_rel[col];
            if (RELU) val = fmaxf(val, 0.0f);
            out[row * COUT + col] = val;
        }
    }
}

// --------------------------------------------------------------- GraphNorm
__global__ void gn_accum_kernel(const float* __restrict__ h,
                                const int* __restrict__ batch,
                                float* __restrict__ cnt,
                                float* __restrict__ sum) {
    int i = blockIdx.x * blockDim.x + threadIdx.x; // over N*64
    if (i >= NNODES * 64) return;
    int n = i >> 6, c = i & 63;
    int g = batch[n];
    atomicAdd(&sum[g * 64 + c], h[i]);
    if (c == 0) atomicAdd(&cnt[g], 1.0f);
}

__global__ void gn_mean_kernel(float* __restrict__ sum,  // in-place -> mean
                               const float* __restrict__ cnt) {
    int i = blockIdx.x * blockDim.x + threadIdx.x; // over G*64
    if (i >= NGRAPH * 64) return;
    sum[i] = sum[i] / cnt[i >> 6];
}

__global__ void gn_subvar_kernel(float* __restrict__ h,            // in-place sub
                                 const int* __restrict__ batch,
                                 const float* __restrict__ mean,   // [G,64]
                                 const float* __restrict__ mscale, // [64]
                                 float* __restrict__ var) {        // [G,64] accum
    int i = blockIdx.x * blockDim.x + threadIdx.x; // over N*64
    if (i >= NNODES * 64) return;
    int n = i >> 6, c = i & 63;
    int g = batch[n];
    float s = h[i] - mean[g * 64 + c] * mscale[c];
    h[i] = s;
    atomicAdd(&var[g * 64 + c], s * s);
}

__global__ void gn_inv_kernel(float* __restrict__ var,  // in-place -> rsqrt
                              const float* __restrict__ cnt) {
    int i = blockIdx.x * blockDim.x + threadIdx.x; // over G*64
    if (i >= NGRAPH * 64) return;
    var[i] = rsqrtf(var[i] / cnt[i >> 6] + GN_EPS);
}

__global__ void gn_apply_kernel(float* __restrict__ h, // in-place normalize
                                const int* __restrict__ batch,
                                const float* __restrict__ inv, // [G,64]
                                const float* __restrict__ w,
                                const float* __restrict__ b) {
    int i = blockIdx.x * blockDim.x + threadIdx.x; // over N*64
    if (i >= NNODES * 64) return;
    int n = i >> 6, c = i & 63;
    int g = batch[n];
    h[i] = w[c] * h[i] * inv[g * 64 + c] + b[c];
}

// ---------------------------------------------------------------- launcher
extern "C" void kernel_launch(void* const* d_in, const int* in_sizes, int n_in,
                              void* d_out, int out_size, void* d_ws, size_t ws_size,
                              hipStream_t stream) {
    const int N = NNODES, E = NEDGES, G = NGRAPH;
    const float* x       = (const float*)d_in[0];
    const int*   ei      = (const int*)d_in[1];
    const int*   src     = ei;
    const int*   dst     = ei + E;
    const float* ew      = (const float*)d_in[2];
    const int*   batch   = (const int*)d_in[3];
    const float* w_rel1  = (const float*)d_in[4];
    const float* b_rel1  = (const float*)d_in[5];
    const float* w_root1 = (const float*)d_in[6];
    const float* w_rel2  = (const float*)d_in[7];
    const float* b_rel2  = (const float*)d_in[8];
    const float* w_root2 = (const float*)d_in[9];
    const float* w_rel3  = (const float*)d_in[10];
    const float* b_rel3  = (const float*)d_in[11];
    const float* w_root3 = (const float*)d_in[12];
    const float* w_relmu = (const float*)d_in[13];
    const float* b_relmu = (const float*)d_in[14];
    const float* w_rootmu= (const float*)d_in[15];
    const float* w_rellv = (const float*)d_in[16];
    const float* b_rellv = (const float*)d_in[17];
    const float* w_rootlv= (const float*)d_in[18];
    const float* gn_w    = (const float*)d_in[19];
    const float* gn_b    = (const float*)d_in[20];
    const float* gn_ms   = (const float*)d_in[21];

    float* mu = (float*)d_out;
    float* lv = (float*)d_out + (size_t)N * 64;

    // workspace layout (fp32): agg[N*64] h1[N*16] h2[N*32] h3[N*64]
    //                          cnt[G] sum[G*64] var[G*64]   (~352 MB)
    float* agg = (float*)d_ws;
    float* h1  = agg + (size_t)N * 64;
    float* h2  = h1  + (size_t)N * 16;
    float* h3  = h2  + (size_t)N * 32;
    float* cnt = h3  + (size_t)N * 64;
    float* sum = cnt + G;
    float* var = sum + G * 64;

    const int B = 256;
    const int ntiles = N / 16;                    // 31250, exact
    const int wblocks = (ntiles * 32 + B - 1) / B;
    auto g1 = [&](int n) { return dim3((n + B - 1) / B); };

    // ---- layer 1: 1 -> 16 -----------------------------------------------
    fill0_kernel<<<g1(N), B, 0, stream>>>(agg, N);
    edge_scatter_kernel<1><<<g1(E), B, 0, stream>>>(x, src, dst, ew, agg, E);
    conv1_node_kernel<<<g1(N * 16), B, 0, stream>>>(agg, x, w_rel1, b_rel1, w_root1, h1);

    // ---- layer 2: 16 -> 32 ----------------------------------------------
    fill0_kernel<<<g1(N * 16), B, 0, stream>>>(agg, N * 16);
    edge_scatter_kernel<16><<<g1(E * 16), B, 0, stream>>>(h1, src, dst, ew, agg, E * 16);
    conv_node_wmma_kernel<16, 32, true><<<wblocks, B, 0, stream>>>(
        agg, h1, w_rel2, w_root2, b_rel2, h2, ntiles);

    // ---- layer 3: 32 -> 64 ----------------------------------------------
    fill0_kernel<<<g1(N * 32), B, 0, stream>>>(agg, N * 32);
    edge_scatter_kernel<32><<<g1(E * 32), B, 0, stream>>>(h2, src, dst, ew, agg, E * 32);
    conv_node_wmma_kernel<32, 64, true><<<wblocks, B, 0, stream>>>(
        agg, h2, w_rel3, w_root3, b_rel3, h3, ntiles);

    // ---- GraphNorm (in place on h3) -------------------------------------
    fill0_kernel<<<g1(G), B, 0, stream>>>(cnt, G);
    fill0_kernel<<<g1(G * 64), B, 0, stream>>>(sum, G * 64);
    fill0_kernel<<<g1(G * 64), B, 0, stream>>>(var, G * 64);
    gn_accum_kernel<<<g1(N * 64), B, 0, stream>>>(h3, batch, cnt, sum);
    gn_mean_kernel<<<g1(G * 64), B, 0, stream>>>(sum, cnt);
    gn_subvar_kernel<<<g1(N * 64), B, 0, stream>>>(h3, batch, sum, gn_ms, var);
    gn_inv_kernel<<<g1(G * 64), B, 0, stream>>>(var, cnt);
    gn_apply_kernel<<<g1(N * 64), B, 0, stream>>>(h3, batch, var, gn_w, gn_b);

    // ---- mu / logvar heads: shared 64-chan gather/scatter ---------------
    fill0_kernel<<<g1(N * 64), B, 0, stream>>>(agg, N * 64);
    edge_scatter_kernel<64><<<g1(E * 64), B, 0, stream>>>(h3, src, dst, ew, agg, E * 64);
    conv_node_wmma_kernel<64, 64, false><<<wblocks, B, 0, stream>>>(
        agg, h3, w_relmu, w_rootmu, b_relmu, mu, ntiles);
    conv_node_wmma_kernel<64, 64, false><<<wblocks, B, 0, stream>>>(
        agg, h3, w_rellv, w_rootlv, b_rellv, lv, ntiles);
}